// GaussianMapping_22832046145900
// MI455X (gfx1250) — compile-verified
//
#include <hip/hip_runtime.h>
#include <cmath>

typedef float v2f __attribute__((ext_vector_type(2)));
typedef float v8f __attribute__((ext_vector_type(8)));

#define D_MODEL 256
#define N_RF    128
#define ROWS    (8 * 8192)       // B*T
#define KC      64               // K chunk staged in LDS
#define NCHUNK  (D_MODEL / KC)
#define UPAD    68               // padded LDS row stride (dwords): 4*r banks -> conflict-free b64 reads

// ---------------------------------------------------------------------------
// Kernel 1: alpha[r] = ||U_r||^2 * (1/q - 2)
// ---------------------------------------------------------------------------
__global__ __launch_bounds__(128) void gm_alpha_kernel(const float* __restrict__ U,
                                                       float* __restrict__ alpha,
                                                       float acoef) {
    const int r = threadIdx.x;                      // 0..127
    const float4* Ur = (const float4*)(U + (size_t)r * D_MODEL);
    float s = 0.f;
#pragma unroll 8
    for (int i = 0; i < D_MODEL / 4; ++i) {
        float4 u = Ur[i];
        s += u.x * u.x + u.y * u.y + u.z * u.z + u.w * u.w;
    }
    alpha[r] = s * acoef;
}

// ---------------------------------------------------------------------------
// Kernel 2: out[m, r] = scale * exp(alpha[r] - 2*x2[m] + 4 * (x @ U^T)[m, r])
// One wave = one 16-row M tile x all 128 columns (8 WMMA N-tiles).
// x is streamed once (NT loads); out written once, never re-read (NT stores).
// U (128 KB) stays hot: staged chunk-wise into LDS with regular-temporal loads.
// ---------------------------------------------------------------------------
__global__ __launch_bounds__(256) void gm_main_kernel(const float* __restrict__ x,
                                                      const float* __restrict__ U,
                                                      const float* __restrict__ alpha,
                                                      float* __restrict__ out,
                                                      float scale) {
    __shared__ float Ulds[N_RF][UPAD];   // ~34 KB, K-chunk of U
    __shared__ float alds[N_RF];

    const int tid  = threadIdx.x;
    const int lane = tid & 31;
    const int wave = tid >> 5;
    const int hi   = lane >> 4;          // half-lane select (K/M split per ISA layouts)
    const int lo   = lane & 15;

    const int rowBase = blockIdx.x * 128 + wave * 16;         // this wave's 16 rows
    const float* xrow = x + (size_t)(rowBase + lo) * D_MODEL; // A-fragment row for this lane

    if (tid < N_RF) alds[tid] = alpha[tid];

    v8f acc[8] = {};     // 8 N-tiles of 16x16 f32 accumulators (64 VGPRs)
    float x2p = 0.f;     // partial sum-of-squares for row `lo` (this half-lane's K slices)

    for (int kc = 0; kc < NCHUNK; ++kc) {
        const int kbase = kc * KC;
        __syncthreads();
        // Stage U[:, kbase .. kbase+63] -> LDS, 128x64 floats as float4 (fully coalesced)
#pragma unroll
        for (int i = 0; i < 8; ++i) {
            int e  = i * 256 + tid;          // float4 index 0..2047
            int r  = e >> 4;                 // U row
            int c4 = (e & 15) << 2;          // col within chunk
            float4 u = *(const float4*)(U + (size_t)r * D_MODEL + kbase + c4);
            *(float4*)(&Ulds[r][c4]) = u;    // UPAD%4==0 -> 16B aligned ds_store_b128
        }
        // Prefetch next x chunk for this lane's row (CDNA5 global_prefetch_b8)
        if (kc + 1 < NCHUNK)
            __builtin_prefetch(xrow + kbase + KC + 2 * hi, 0, 1);
        __syncthreads();

        // 16 K-steps of 4: A fragment = contiguous float2 per lane (ISA 16x4 f32 A layout)
#pragma unroll
        for (int kt = 0; kt < KC / 4; ++kt) {
            const int cfrag = kt * 4 + 2 * hi;
            // x element is touched exactly once across the whole kernel -> TH=NT load
            v2f a = __builtin_nontemporal_load((const v2f*)(xrow + kbase + cfrag));
            x2p += a.x * a.x + a.y * a.y;
#pragma unroll
            for (int nt = 0; nt < 8; ++nt) {
                v2f b = *(const v2f*)(&Ulds[nt * 16 + lo][cfrag]);  // ds_load_b64, conflict-free
                acc[nt] = __builtin_amdgcn_wmma_f32_16x16x4_f32(
                    false, a, false, b, (short)0, acc[nt], false, false);
            }
        }
    }

    // Full ||x_row||^2: combine the two half-lane K partials (wave32 shuffle)
    const float x2 = x2p + __shfl_xor(x2p, 16, 32);

    // Epilogue: C/D layout -> VGPR v holds M = v + 8*hi, N = lo
#pragma unroll
    for (int nt = 0; nt < 8; ++nt) {
        const float ar = alds[nt * 16 + lo];
#pragma unroll
        for (int v = 0; v < 8; ++v) {
            const int   mloc = v + 8 * hi;
            const float xm   = __shfl(x2, mloc, 32);   // x2 for row mloc lives in lane mloc
            const float val  = scale * __expf(ar - 2.f * xm + 4.f * acc[nt][v]);
            // output is write-once, never re-read by the GPU -> TH=NT store
            __builtin_nontemporal_store(
                val, out + (size_t)(rowBase + mloc) * N_RF + nt * 16 + lo);
        }
    }
}

// ---------------------------------------------------------------------------
// Host launcher
// ---------------------------------------------------------------------------
extern "C" void kernel_launch(void* const* d_in, const int* in_sizes, int n_in,
                              void* d_out, int out_size, void* d_ws, size_t ws_size,
                              hipStream_t stream) {
    const float* x   = (const float*)d_in[0];   // (8, 8192, 256) f32
    const float* U   = (const float*)d_in[1];   // (128, 256) f32
    float*       out = (float*)d_out;           // (8, 8192, 128) f32
    float*       alp = (float*)d_ws;            // 128 floats scratch

    // q = exp(W(y))/2 with y = R^2/(REG*D) = 1/256 ; Newton on Lambert W (as reference)
    const double y = 1.0 / 256.0;
    double w = log1p(y);
    for (int i = 0; i < 50; ++i) {
        double ew = exp(w);
        w = w - (w * ew - y) / (ew * (w + 1.0));
    }
    const double q = exp(w) / 2.0;
    const double C = pow(2.0 * q, (double)D_MODEL / 4.0);   // (2q)^64
    const float scale = (float)(C / sqrt((double)N_RF));
    const float acoef = (float)(1.0 / q - 2.0);             // alpha coefficient on u2

    gm_alpha_kernel<<<1, 128, 0, stream>>>(U, alp, acoef);
    gm_main_kernel<<<ROWS / 128, 256, 0, stream>>>(x, U, alp, out, scale);
}